// LightGCN_23759759082206
// MI455X (gfx1250) — compile-verified
//
#include <hip/hip_runtime.h>

#define NN 100000
#define DD 128
#define EE 3200000
#define DV4 (DD / 4)      // 32 float4 per row == one wave32, one float4 per lane
#define EB  (EE / 32)     // 100000 batches of 32 edges (exact, no tail)

// ================= edge scatter: one wave32 per 32-edge batch =================
// lane l loads edge e0+l's endpoints (coalesced b32), then the wave iterates the
// 32 edges with r/c/nrm broadcast into SGPRs via v_readlane. The source feature
// row (512B) is staged global->LDS with GLOBAL_LOAD_ASYNC_TO_LDS_B128 through a
// per-wave double buffer (ASYNCcnt pipeline, one row in flight), then read back
// with ds_load_b128 and scatter-added with global_atomic_add_f32. Next batch's
// source rows are prefetched into L2 a full batch ahead (global_prefetch_b8).

__global__ void k_edges(const float* __restrict__ hin,
                        float* __restrict__ hout,
                        const int* __restrict__ row,
                        const int* __restrict__ col,
                        const float* __restrict__ dis) {
    __shared__ float4 stage[8][2][32];            // 8 waves x 2 bufs x 512B = 8KB
    const int lane = threadIdx.x & 31;
    const int w    = threadIdx.x >> 5;
    const int wid  = blockIdx.x * (blockDim.x >> 5) + w;
    const int nw   = gridDim.x * (blockDim.x >> 5);

    // per-lane LDS byte addresses of the two staging slots (low 32 bits of the
    // generic pointer are the LDS offset)
    const unsigned lds0 = (unsigned)(uintptr_t)&stage[w][0][lane];
    const unsigned lds1 = (unsigned)(uintptr_t)&stage[w][1][lane];

    for (int b = wid; b < EB; b += nw) {
        const int e = b * 32 + lane;
        int   r   = row[e];
        int   c   = col[e];
        float nrm = dis[r] * dis[c];

        // prefetch next batch's source rows into L2 (512B = 4 lines per row)
        const int bn = b + nw;
        if (bn < EB) {
            int rn = row[bn * 32 + lane];
            const char* p = (const char*)(hin + rn * DD);
            __builtin_prefetch(p,       0, 3);
            __builtin_prefetch(p + 128, 0, 3);
            __builtin_prefetch(p + 256, 0, 3);
            __builtin_prefetch(p + 384, 0, 3);
        }

        // preload edge 0's source row into buffer 0
        {
            const int r0 = __builtin_amdgcn_readlane(r, 0);
            const float4* src = (const float4*)(hin + r0 * DD) + lane;
            asm volatile("global_load_async_to_lds_b128 %0, %1, off"
                         :: "v"(lds0), "v"(src) : "memory");
        }

        for (int j = 0; j < 32; ++j) {
            // issue async stage of edge j+1's row into the other buffer, then
            // wait until edge j's row (older async, in-order) has landed
            if (j < 31) {
                const int rn = __builtin_amdgcn_readlane(r, j + 1);
                const unsigned nxt = (j & 1) ? lds0 : lds1;
                const float4* src = (const float4*)(hin + rn * DD) + lane;
                asm volatile("global_load_async_to_lds_b128 %0, %1, off"
                             :: "v"(nxt), "v"(src) : "memory");
                asm volatile("s_wait_asynccnt 0x1" ::: "memory");
            } else {
                asm volatile("s_wait_asynccnt 0x0" ::: "memory");
            }

            const int   cj = __builtin_amdgcn_readlane(c, j);             // SGPR
            const float nj = __uint_as_float(
                __builtin_amdgcn_readlane(__float_as_uint(nrm), j));      // SGPR

            // read the staged row back (ds_load_b128)
            const float4 v = stage[w][j & 1][lane];

            float* dst = hout + cj * DD + lane * 4;
            unsafeAtomicAdd(dst + 0, nj * v.x);   // global_atomic_add_f32
            unsafeAtomicAdd(dst + 1, nj * v.y);
            unsafeAtomicAdd(dst + 2, nj * v.z);
            unsafeAtomicAdd(dst + 3, nj * v.w);
        }
    }
}

// ================= degree / normalization =================

__global__ void k_init_deg(int* __restrict__ deg) {
    int i = blockIdx.x * blockDim.x + threadIdx.x;
    if (i < NN) deg[i] = 1;                       // self loop contributes 1
}

__global__ void k_count_deg(const int* __restrict__ col, int* __restrict__ deg) {
    int e = blockIdx.x * blockDim.x + threadIdx.x;
    if (e < EE) atomicAdd(&deg[col[e]], 1);
}

__global__ void k_rsqrt(const int* __restrict__ deg, float* __restrict__ dis) {
    int i = blockIdx.x * blockDim.x + threadIdx.x;
    if (i < NN) dis[i] = 1.0f / sqrtf((float)deg[i]);   // deg >= 1, never inf
}

// ============ init: out = x, h = dis^2 * x (self-loop of layer 0) ============

__global__ void k_init(const float4* __restrict__ x4,
                       float4* __restrict__ out4,
                       float4* __restrict__ h4,
                       const float* __restrict__ dis) {
    int i = blockIdx.x * blockDim.x + threadIdx.x;
    if (i < NN * DV4) {
        float d = dis[i >> 5];
        float s = d * d;
        float4 v = x4[i];
        out4[i] = v;
        float4 w = {s * v.x, s * v.y, s * v.z, s * v.w};
        h4[i] = w;
    }
}

// ==== fused: out += h_prev ; h_next = dis^2 * h_prev (self-loop next layer) ====

__global__ void k_fuse(const float4* __restrict__ hprev4,
                       float4* __restrict__ out4,
                       float4* __restrict__ hnext4,
                       const float* __restrict__ dis) {
    int i = blockIdx.x * blockDim.x + threadIdx.x;
    if (i < NN * DV4) {
        float d = dis[i >> 5];
        float s = d * d;
        float4 h = hprev4[i];
        float4 o = out4[i];
        o.x += h.x; o.y += h.y; o.z += h.z; o.w += h.w;
        out4[i] = o;
        float4 w = {s * h.x, s * h.y, s * h.z, s * h.w};
        hnext4[i] = w;
    }
}

// ================= final: out = (out + h_last) * 0.25 =================

__global__ void k_final(float4* __restrict__ out4,
                        const float4* __restrict__ h4) {
    int i = blockIdx.x * blockDim.x + threadIdx.x;
    if (i < NN * DV4) {
        float4 h = h4[i];
        float4 o = out4[i];
        o.x = (o.x + h.x) * 0.25f;
        o.y = (o.y + h.y) * 0.25f;
        o.z = (o.z + h.z) * 0.25f;
        o.w = (o.w + h.w) * 0.25f;
        out4[i] = o;
    }
}

// ================= launcher =================

extern "C" void kernel_launch(void* const* d_in, const int* in_sizes, int n_in,
                              void* d_out, int out_size, void* d_ws, size_t ws_size,
                              hipStream_t stream) {
    const float* x  = (const float*)d_in[0];      // [NN, DD]
    const int*   ei = (const int*)d_in[1];        // [2, EE]
    const int* row = ei;
    const int* col = ei + EE;
    float* out = (float*)d_out;

    // workspace: deg[NN] int | dis[NN] f32 | hA[NN*DD] | hB[NN*DD]  (~103 MB)
    char* ws = (char*)d_ws;
    int*   deg = (int*)  (ws);
    float* dis = (float*)(ws + (size_t)NN * 4);
    float* hA  = (float*)(ws + (size_t)NN * 8);
    float* hB  = (float*)(ws + (size_t)NN * 8 + (size_t)NN * DD * 4);

    const int B = 256;
    const int gN    = (NN + B - 1) / B;
    const int gE    = (EE + B - 1) / B;
    const int gND4  = (NN * DV4 + B - 1) / B;     // 3.2M threads, float4 sweeps
    const int gEdge = 2048;                       // 16384 waves, ~6 batches each

    // normalization: deg (with self loop) -> dis = deg^-1/2
    k_init_deg <<<gN, B, 0, stream>>>(deg);
    k_count_deg<<<gE, B, 0, stream>>>(col, deg);
    k_rsqrt    <<<gN, B, 0, stream>>>(deg, dis);

    // layer 0: out = x ; hA = dis^2*x ; hA += edges(x)
    k_init <<<gND4, B, 0, stream>>>((const float4*)x, (float4*)out, (float4*)hA, dis);
    k_edges<<<gEdge, B, 0, stream>>>(x, hA, row, col, dis);

    // layer 1: out += hA ; hB = dis^2*hA ; hB += edges(hA)
    k_fuse <<<gND4, B, 0, stream>>>((const float4*)hA, (float4*)out, (float4*)hB, dis);
    k_edges<<<gEdge, B, 0, stream>>>(hA, hB, row, col, dis);

    // layer 2: out += hB ; hA = dis^2*hB ; hA += edges(hB)
    k_fuse <<<gND4, B, 0, stream>>>((const float4*)hB, (float4*)out, (float4*)hA, dis);
    k_edges<<<gEdge, B, 0, stream>>>(hB, hA, row, col, dis);

    // out = (out + hA) / 4
    k_final<<<gND4, B, 0, stream>>>((float4*)out, (const float4*)hA);
}